// InterLayerTrajectoryFlow_43224550867592
// MI455X (gfx1250) — compile-verified
//
#include <hip/hip_runtime.h>
#include <math.h>

// Problem constants (from reference setup_inputs)
#define BB      8
#define SS      8192
#define DD      512
#define WINDOW  6

// Kernel tiling
#define TS      32      // positions per block along S
#define NT      128     // threads per block (4 x wave32); NT*4 == DD
#define PF      4       // async rows in flight
#define RING    8       // LDS ring slots (power of 2: % -> AND; reuse distance 8 > PF)

typedef int   v4i __attribute__((vector_size(16)));
typedef float v4f __attribute__((vector_size(16)));

#if __has_builtin(__builtin_amdgcn_global_load_async_to_lds_b128)
#define HAVE_ASYNC 1
#else
#define HAVE_ASYNC 0
#endif

#if HAVE_ASYNC
typedef __attribute__((address_space(1))) v4i gv4i_t;
typedef __attribute__((address_space(3))) v4i lv4i_t;
#endif

__device__ __forceinline__ void stage_row16(const float* g, float4* l) {
#if HAVE_ASYNC
  // GLOBAL_LOAD_ASYNC_TO_LDS_B128: 16B per lane, tracked with ASYNCcnt
  __builtin_amdgcn_global_load_async_to_lds_b128((gv4i_t*)g, (lv4i_t*)l, 0, 0);
#else
  *l = *(const float4*)g;
#endif
}

__device__ __forceinline__ void wait_ring() {
#if HAVE_ASYNC
#if __has_builtin(__builtin_amdgcn_s_wait_asynccnt)
  __builtin_amdgcn_s_wait_asynccnt(PF - 1);
#else
  asm volatile("s_wait_asynccnt 3" ::: "memory");
#endif
#endif
}

__global__ __launch_bounds__(NT) void traj_flow_kernel(
    const float* __restrict__ emb,
    const int*   __restrict__ lidx,
    float*       __restrict__ out)
{
  __shared__ float4 ring[RING][NT];   // 16 KB row ring
  __shared__ float  red[2][4];        // 2-phase cross-wave reduction scratch

  const int tid = threadIdx.x;
  const int wid = tid >> 5;
  const int b   = blockIdx.y;
  const int s0  = blockIdx.x * TS;
  const int s1  = s0 + TS;
  const int t0  = (s0 >= WINDOW) ? (s0 - WINDOW) : 0;

  const float* ebase = emb + ((size_t)b * SS) * DD + (size_t)tid * 4;
  float*       obase = out + ((size_t)b * SS) * DD + (size_t)tid * 4;

  // constant weights for full window (w == 6): exp(-k/5)
  float Wc0 = 1.0f;
  float Wc1 = expf(-0.2f);
  float Wc2 = expf(-0.4f);
  float Wc3 = expf(-0.6f);
  float Wc4 = expf(-0.8f);
  float Wc5 = expf(-1.0f);
  const float wsum6 = Wc0 + Wc1 + Wc2 + Wc3 + Wc4 + Wc5;
  const float depth = 0.1f * (1.0f + (float)(*lidx) * 0.8f);
  const float m6    = depth / (wsum6 + 1e-8f);

  const float4 z = make_float4(0.f, 0.f, 0.f, 0.f);
  float4 n0 = z, n1 = z, n2 = z, n3 = z, n4 = z, n5 = z;  // newest..oldest diffs
  float4 eprev = z;
  int phase = 0;
  int nextr = t0;

  // fill the async pipeline: rows t0 .. t0+PF-1 (always valid: >= TS rows per block)
#pragma unroll
  for (int i = 0; i < PF; ++i) {
    stage_row16(ebase + (size_t)nextr * DD, &ring[nextr & (RING - 1)][tid]);
    ++nextr;
  }

  // consume row r from LDS ring, keep pipeline depth constant (tail issues are
  // clamped re-reads of the last row -> L2 hits, keeps wait immediate constant)
  auto consume = [&](int r) -> float4 {
    wait_ring();                              // oldest of PF in-flight is done
    float4 v = ring[r & (RING - 1)][tid];
    int rr = (nextr < (SS - 1)) ? nextr : (SS - 1);
    stage_row16(ebase + (size_t)rr * DD, &ring[nextr & (RING - 1)][tid]);
    ++nextr;
    return v;
  };

  // consume row r, producing normalized diff t = r-1 into the register window
  auto push_diff = [&](int r) {
    float4 v = consume(r);
    float dx = v.x - eprev.x, dy = v.y - eprev.y;
    float dz = v.z - eprev.z, dw = v.w - eprev.w;
    eprev = v;
    float p = dx * dx + dy * dy + dz * dz + dw * dw;
#pragma unroll
    for (int off = 16; off > 0; off >>= 1) p += __shfl_xor(p, off, 32);
    if ((tid & 31) == 0) red[phase][wid] = p;
    __syncthreads();
    float s2 = red[phase][0] + red[phase][1] + red[phase][2] + red[phase][3];
    phase ^= 1;
    float mag = sqrtf(s2);
    float inv = (mag > 1e-6f) ? (1.0f / (mag + 1e-8f)) : 0.0f;
    n5 = n4; n4 = n3; n3 = n2; n2 = n1; n1 = n0;
    n0.x = dx * inv; n0.y = dy * inv; n0.z = dz * inv; n0.w = dw * inv;
  };

  // warm-up: eprev = row t0; diffs t0 .. s0-2
  eprev = consume(t0);
  for (int r = t0 + 1; r < s0; ++r) push_diff(r);

  for (int pos = s0; pos < s1; ++pos) {
    if (pos >= 1) push_diff(pos);   // diff t = pos-1 becomes n0

    float ax = 0.f, ay = 0.f, az = 0.f, aw = 0.f;
    float mult;
    if (pos >= WINDOW) {
      ax = Wc0*n0.x + Wc1*n1.x + Wc2*n2.x + Wc3*n3.x + Wc4*n4.x + Wc5*n5.x;
      ay = Wc0*n0.y + Wc1*n1.y + Wc2*n2.y + Wc3*n3.y + Wc4*n4.y + Wc5*n5.y;
      az = Wc0*n0.z + Wc1*n1.z + Wc2*n2.z + Wc3*n3.z + Wc4*n4.z + Wc5*n5.z;
      aw = Wc0*n0.w + Wc1*n1.w + Wc2*n2.w + Wc3*n3.w + Wc4*n4.w + Wc5*n5.w;
      mult = m6;
    } else {
      // only the first S-tile (pos 0..5) takes this path
      const int w = pos;                 // w = min(pos, 6) == pos here
      float wsum = 0.f;
      const float den = (w > 1) ? (float)(w - 1) : 1.0f;
      if (w > 0) { float t = (w == 1) ? expf(-1.0f) : 1.0f;
                   ax += t*n0.x; ay += t*n0.y; az += t*n0.z; aw += t*n0.w; wsum += t; }
      if (w > 1) { float t = expf(-1.0f / den);
                   ax += t*n1.x; ay += t*n1.y; az += t*n1.z; aw += t*n1.w; wsum += t; }
      if (w > 2) { float t = expf(-2.0f / den);
                   ax += t*n2.x; ay += t*n2.y; az += t*n2.z; aw += t*n2.w; wsum += t; }
      if (w > 3) { float t = expf(-3.0f / den);
                   ax += t*n3.x; ay += t*n3.y; az += t*n3.z; aw += t*n3.w; wsum += t; }
      if (w > 4) { float t = expf(-4.0f / den);
                   ax += t*n4.x; ay += t*n4.y; az += t*n4.z; aw += t*n4.w; wsum += t; }
      mult = depth / (wsum + 1e-8f);
    }

    // output is write-once: NT store keeps the 128MB output stream from evicting
    // the input working set out of the 192MB L2 (halo rows between tiles hit L2)
    v4f o = { ax * mult, ay * mult, az * mult, aw * mult };
    __builtin_nontemporal_store(o, (v4f*)(obase + (size_t)pos * DD));
  }
}

extern "C" void kernel_launch(void* const* d_in, const int* in_sizes, int n_in,
                              void* d_out, int out_size, void* d_ws, size_t ws_size,
                              hipStream_t stream) {
  (void)in_sizes; (void)n_in; (void)out_size; (void)d_ws; (void)ws_size;
  const float* emb  = (const float*)d_in[0];
  const int*   lidx = (const int*)d_in[1];
  float*       out  = (float*)d_out;

  dim3 grid(SS / TS, BB, 1);   // 256 x 8 = 2048 workgroups
  dim3 block(NT, 1, 1);        // 4 wave32 per workgroup
  hipLaunchKernelGGL(traj_flow_kernel, grid, block, 0, stream, emb, lidx, out);
}